// GivensRotationLayer_38104949850705
// MI455X (gfx1250) — compile-verified
//
#include <hip/hip_runtime.h>

// ---------------------------------------------------------------------------
// GivensRotationLayer for MI455X (gfx1250, wave32)
//   Phase 1: build U (512x512 f32) row-parallel in LDS, 511 sequential rounds.
//   Phase 2: out = x @ U^T + bias via V_WMMA_F32_16X16X4_F32 (full f32).
// Memory-bound problem (~65MB min traffic -> ~2.8us @ 23.3TB/s), so f32 WMMA
// keeps reference precision at zero perf cost.
// ---------------------------------------------------------------------------

typedef __attribute__((ext_vector_type(2))) float v2f;
typedef __attribute__((ext_vector_type(8))) float v8f;

#define NDIM   512
#define NBATCH 16384
#define NROUND 511
#define NPAIR  256

// ============================ Phase 1: build U =============================
// Each block owns ROWS=8 rows of U in LDS, laid out [col][row] (rows[c*8+r]).
// Thread p handles pair p of every round for all 8 rows. Pairs within a round
// are disjoint (round-robin schedule), so only one barrier per round.
#define ROWS 8

__global__ __launch_bounds__(256) void build_u_kernel(
    const float* __restrict__ angles,   // [511*256]
    const int*   __restrict__ blocks,   // [511*256*2]
    float*       __restrict__ U)        // [512*512] row-major
{
    __shared__ float rows[NDIM * ROWS];   // 16 KB
    const int t  = threadIdx.x;           // 0..255
    const int r0 = blockIdx.x * ROWS;     // first global row of this block

    // init: row (r0+rr) = e_{r0+rr}
    for (int idx = t; idx < NDIM * ROWS; idx += 256) {
        int c  = idx >> 3;
        int rr = idx & 7;
        rows[idx] = (c == r0 + rr) ? 1.0f : 0.0f;
    }
    __syncthreads();

    for (int rd = 0; rd < NROUND; ++rd) {
        const int   base = rd * NPAIR + t;
        const int   i    = blocks[2 * base + 0];
        const int   j    = blocks[2 * base + 1];
        const float a    = angles[base];
        const float cth  = cosf(a);
        const float sth  = sinf(a);
        float* __restrict__ pi = &rows[i << 3];
        float* __restrict__ pj = &rows[j << 3];
        #pragma unroll
        for (int rr = 0; rr < ROWS; ++rr) {
            const float ui = pi[rr];
            const float uj = pj[rr];
            pi[rr] = cth * ui + sth * uj;   // U[:, i] = c*Ui + s*Uj
            pj[rr] = cth * uj - sth * ui;   // U[:, j] = -s*Ui + c*Uj
        }
        __syncthreads();
    }

    // write rows to global: contiguous in c across threads
    for (int idx = t; idx < NDIM * ROWS; idx += 256) {
        const int c  = idx & (NDIM - 1);
        const int rr = idx >> 9;
        U[(r0 + rr) * NDIM + c] = rows[(c << 3) + rr];
    }
}

// ====================== Phase 2: out = x @ U^T + bias ======================
// Workgroup tile: BM=128 x BN=64, K staged in LDS chunks of BK=64.
// 8 waves arranged 4(M) x 2(N); each wave computes a 32x32 sub-tile as
// 2x2 accumulators of v_wmma_f32_16x16x4_f32.
#define BM  128
#define BN  64
#define BK  64
#define LDK 68   // padded row stride (floats): stride-68 float2 reads hit all 64 banks

__global__ __launch_bounds__(256) void givens_gemm_kernel(
    const float* __restrict__ x,     // [16384 x 512]
    const float* __restrict__ U,     // [512 x 512]
    const float* __restrict__ bias,  // [512]
    float*       __restrict__ out)   // [16384 x 512]
{
    __shared__ float sx[BM * LDK];   // 34816 B : x tile  [m][k]
    __shared__ float su[BN * LDK];   // 17408 B : U tile  [n][k]  (== B = U^T in [k][n] sense)

    const int t    = threadIdx.x;
    const int lane = t & 31;
    const int wave = t >> 5;          // 0..7
    const int lm   = lane & 15;       // lane within half
    const int half = lane >> 4;       // 0/1 -> K pair {0,1} vs {2,3}
    const int wm   = wave & 3;        // 4 waves along M
    const int wn   = wave >> 2;       // 2 waves along N

    const int m0 = blockIdx.x * BM;
    const int n0 = blockIdx.y * BN;

    v8f acc[2][2] = {};

    for (int kc = 0; kc < NDIM; kc += BK) {
        // ---- cooperative global -> LDS (float4, fully coalesced) ----
        {
            const int c4 = (t & 15) << 2;               // 0,4,...,60
            const int rb = t >> 4;                      // 0..15
            #pragma unroll
            for (int r = rb; r < BM; r += 16) {
                const float4 v = *(const float4*)&x[(m0 + r) * NDIM + kc + c4];
                *(float4*)&sx[r * LDK + c4] = v;
            }
            #pragma unroll
            for (int r = rb; r < BN; r += 16) {
                const float4 v = *(const float4*)&U[(n0 + r) * NDIM + kc + c4];
                *(float4*)&su[r * LDK + c4] = v;
            }
        }
        __syncthreads();

        // ---- WMMA over this K chunk ----
        const int arow0 = (wm * 32 +      lm) * LDK + 2 * half;
        const int arow1 = (wm * 32 + 16 + lm) * LDK + 2 * half;
        const int brow0 = (wn * 32 +      lm) * LDK + 2 * half;
        const int brow1 = (wn * 32 + 16 + lm) * LDK + 2 * half;
        #pragma unroll
        for (int kk = 0; kk < BK; kk += 4) {
            const v2f a0 = *(const v2f*)&sx[arow0 + kk];
            const v2f a1 = *(const v2f*)&sx[arow1 + kk];
            const v2f b0 = *(const v2f*)&su[brow0 + kk];
            const v2f b1 = *(const v2f*)&su[brow1 + kk];
            acc[0][0] = __builtin_amdgcn_wmma_f32_16x16x4_f32(
                false, a0, false, b0, (short)0, acc[0][0], false, false);
            acc[0][1] = __builtin_amdgcn_wmma_f32_16x16x4_f32(
                false, a0, false, b1, (short)0, acc[0][1], false, false);
            acc[1][0] = __builtin_amdgcn_wmma_f32_16x16x4_f32(
                false, a1, false, b0, (short)0, acc[1][0], false, false);
            acc[1][1] = __builtin_amdgcn_wmma_f32_16x16x4_f32(
                false, a1, false, b1, (short)0, acc[1][1], false, false);
        }
        __syncthreads();
    }

    // ---- store: C/D layout: VGPR v -> M = v + 8*half, lane -> N = lm ----
    #pragma unroll
    for (int ni = 0; ni < 2; ++ni) {
        const int n  = n0 + wn * 32 + ni * 16 + lm;
        const float bv = bias[n];
        #pragma unroll
        for (int mi = 0; mi < 2; ++mi) {
            const int mbase = m0 + wm * 32 + mi * 16 + half * 8;
            #pragma unroll
            for (int v = 0; v < 8; ++v) {
                out[(mbase + v) * NDIM + n] = acc[mi][ni][v] + bv;
            }
        }
    }
}

// ================================ launcher =================================
extern "C" void kernel_launch(void* const* d_in, const int* in_sizes, int n_in,
                              void* d_out, int out_size, void* d_ws, size_t ws_size,
                              hipStream_t stream) {
    const float* x      = (const float*)d_in[0];   // [16384*512] f32
    const float* angles = (const float*)d_in[1];   // [511*256]   f32
    const float* bias   = (const float*)d_in[2];   // [512]       f32
    const int*   blocks = (const int*)  d_in[3];   // [511*256*2] i32

    float* U   = (float*)d_ws;    // 512*512*4 = 1 MB scratch
    float* out = (float*)d_out;   // [16384*512] f32

    build_u_kernel<<<dim3(NDIM / ROWS), dim3(256), 0, stream>>>(angles, blocks, U);
    givens_gemm_kernel<<<dim3(NBATCH / BM, NDIM / BN), dim3(256), 0, stream>>>(
        x, U, bias, out);
}